// UniGINLayer_7198365188795
// MI455X (gfx1250) — compile-verified
//
#include <hip/hip_runtime.h>

typedef float v2f __attribute__((ext_vector_type(2)));
typedef float v8f __attribute__((ext_vector_type(8)));

#define CCH 128

// One wave32 per nonzero: lane l handles channels [4l, 4l+4).
// Gather a float4 from src[g*128 + 4l], atomic-add into dst[s*128 + 4l].
// Destinations (25.6MB / 51.2MB) are L2-resident on MI455X (192MB L2),
// so the atomics resolve in L2.
__global__ __launch_bounds__(256) void scatter_add_rows(
    const float* __restrict__ src, const int* __restrict__ gather_idx,
    const int* __restrict__ scatter_idx, float* __restrict__ dst, int nnz)
{
    const int t = blockIdx.x * 256 + threadIdx.x;
    const int i = t >> 5;
    if (i >= nnz) return;
    const int lane = t & 31;
    const int g = gather_idx[i];
    const int s = scatter_idx[i];
    const float4 v = *(const float4*)(src + (size_t)g * CCH + lane * 4);
    float* d = dst + (size_t)s * CCH + lane * 4;
    atomicAdd(d + 0, v.x);
    atomicAdd(d + 1, v.y);
    atomicAdd(d + 2, v.z);
    atomicAdd(d + 3, v.w);
}

// out = (x0 + m) @ W^T + b via V_WMMA_F32_16X16X4_F32.
// Block: 256 threads = 8 waves. Block owns 16 M-rows (A tile in LDS, fused add).
// Wave w owns N-tile [16w, 16w+16): B panel (16 rows of W, full K) in registers.
__global__ __launch_bounds__(256) void gin_gemm_wmma(
    const float* __restrict__ x0, const float* __restrict__ mbuf,
    const float* __restrict__ W, const float* __restrict__ bias,
    float* __restrict__ out)
{
    __shared__ float sA[16 * 132];            // padded: bank = 4*m + k, conflict-free

    const int tid = threadIdx.x;
    const int m0  = blockIdx.x * 16;

    // Stage A tile = x0 + m (coalesced; shared by all 8 waves)
    #pragma unroll
    for (int idx = tid; idx < 16 * CCH; idx += 256) {
        const int r = idx >> 7, k = idx & 127;
        const size_t g = (size_t)(m0 + r) * CCH + k;
        sA[r * 132 + k] = x0[g] + mbuf[g];
    }
    __syncthreads();

    const int lane = tid & 31;
    const int wave = tid >> 5;
    const int half = lane >> 4;               // 0: K even pair, 1: K odd pair
    const int lm   = lane & 15;
    const int n    = wave * 16 + lm;          // this lane's fixed output column
    const float bv = bias[n];

    // B fragments: B[k][n] = W[n][k]. VGPR0 holds K=k0+2*half, VGPR1 holds +1.
    v2f bf[32];
    #pragma unroll
    for (int s = 0; s < 32; ++s) {
        const int kk = s * 4 + half * 2;
        bf[s].x = W[(size_t)n * CCH + kk];
        bf[s].y = W[(size_t)n * CCH + kk + 1];
    }

    v8f c = {0.f, 0.f, 0.f, 0.f, 0.f, 0.f, 0.f, 0.f};
    #pragma unroll
    for (int s = 0; s < 32; ++s) {
        const int kk = s * 4 + half * 2;
        v2f a;
        a.x = sA[lm * 132 + kk];
        a.y = sA[lm * 132 + kk + 1];
        c = __builtin_amdgcn_wmma_f32_16x16x4_f32(
                /*neg_a=*/false, a, /*neg_b=*/false, bf[s],
                /*c_mod=*/(short)0, c, /*reuse_a=*/false, /*reuse_b=*/false);
    }

    // D layout: lane half selects rows v (+8), column = n.
    #pragma unroll
    for (int v = 0; v < 8; ++v) {
        const int m = m0 + v + half * 8;
        out[(size_t)m * CCH + n] = c[v] + bv;
    }
}

extern "C" void kernel_launch(void* const* d_in, const int* in_sizes, int n_in,
                              void* d_out, int out_size, void* d_ws, size_t ws_size,
                              hipStream_t stream) {
    const float* x0       = (const float*)d_in[0];
    const int*   node_idx = (const int*)d_in[1];
    const int*   edge_idx = (const int*)d_in[2];
    const float* W        = (const float*)d_in[3];
    const float* bias     = (const float*)d_in[4];

    const int n_nodes = in_sizes[0] / CCH;          // 100000
    const int nnz     = in_sizes[1];                // 1600000
    const int n_edges = out_size / CCH - n_nodes;   // 50000

    float* out0 = (float*)d_out;
    float* x1   = out0 + (size_t)n_nodes * CCH;     // second output lives in d_out
    float* mbuf = (float*)d_ws;                     // m_{1->0} scratch (51.2 MB)

    hipMemsetAsync(x1,   0, (size_t)n_edges * CCH * sizeof(float), stream);
    hipMemsetAsync(mbuf, 0, (size_t)n_nodes * CCH * sizeof(float), stream);

    const long scatter_threads = (long)nnz * 32;
    const int  blk  = 256;
    const long grid = (scatter_threads + blk - 1) / blk;   // 200000 blocks

    // x_1 = B1^T x_0
    scatter_add_rows<<<grid, blk, 0, stream>>>(x0, node_idx, edge_idx, x1, nnz);
    // m = B1 x_1
    scatter_add_rows<<<grid, blk, 0, stream>>>(x1, edge_idx, node_idx, mbuf, nnz);
    // out0 = (x0 + m) @ W^T + b
    gin_gemm_wmma<<<n_nodes / 16, 256, 0, stream>>>(x0, mbuf, W, bias, out0);
}